// GCN_83932250898476
// MI455X (gfx1250) — compile-verified
//
#include <hip/hip_runtime.h>
#include <hip/hip_bf16.h>

#define N_NODES 100000
#define N_EDGES 1600000
#define IN_CH   128
#define HID_CH  128
#define OUT_CH  64

typedef float v2f __attribute__((ext_vector_type(2)));
typedef float v8f __attribute__((ext_vector_type(8)));

// ---------------------------------------------------------------------------
// Degree / normalization kernels
// ---------------------------------------------------------------------------
__global__ __launch_bounds__(256) void gcn_deg_init(float* __restrict__ deg, int n) {
    int i = blockIdx.x * blockDim.x + threadIdx.x;
    if (i < n) deg[i] = 1.0f;   // self-loop contributes 1 to every node's degree
}

__global__ __launch_bounds__(256) void gcn_deg_edges(const long long* __restrict__ dst,
                                                     float* __restrict__ deg, int e) {
    int i = blockIdx.x * blockDim.x + threadIdx.x;
    if (i < e) atomicAdd(&deg[(int)dst[i]], 1.0f);
}

__global__ __launch_bounds__(256) void gcn_deg_to_dinv(float* __restrict__ deg, int n) {
    int i = blockIdx.x * blockDim.x + threadIdx.x;
    if (i < n) deg[i] = rsqrtf(deg[i]);   // deg >= 1 always (self-loops)
}

// ---------------------------------------------------------------------------
// FP32 WMMA GEMM: C[M,N] = A[M,K] * B[K,N]
// One wave per 16x16 output tile, V_WMMA_F32_16X16X4_F32, K stepped by 4.
// Block = 128 threads = 4 waves; wave w handles column tile blockIdx.y*4 + w.
// M, K, N all multiples of 16/4 here, no bounds checks needed.
// ---------------------------------------------------------------------------
__global__ __launch_bounds__(128) void gcn_gemm_wmma_f32(
    const float* __restrict__ A, const float* __restrict__ B,
    float* __restrict__ C, int M, int K, int N) {
    const int lane  = threadIdx.x & 31;
    const int wave  = threadIdx.x >> 5;
    const int tileM = blockIdx.x;
    const int tileN = blockIdx.y * 4 + wave;
    if (tileN * 16 >= N) return;

    const int m    = lane & 15;     // M index for A frag; N index for B/C frags
    const int half = lane >> 4;     // selects K pair (A/B), +8 rows (C/D)

    const float* Arow = A + (size_t)(tileM * 16 + m) * K;
    const float* Bcol = B + (tileN * 16 + m);   // column (N) fixed per lane

    v8f acc = {};
    for (int k = 0; k < K; k += 4) {
        // A 16x4 layout: lanes 0-15 -> K = k,k+1 ; lanes 16-31 -> K = k+2,k+3
        v2f a = *(const v2f*)(Arow + k + half * 2);
        // B 4x16 layout: VGPR0 -> K = k (lanes 0-15) / k+2 (lanes 16-31); VGPR1 -> +1
        v2f b;
        b.x = Bcol[(size_t)(k + half * 2 + 0) * N];
        b.y = Bcol[(size_t)(k + half * 2 + 1) * N];
        acc = __builtin_amdgcn_wmma_f32_16x16x4_f32(
            /*neg_a=*/false, a, /*neg_b=*/false, b,
            /*c_mod=*/(short)0, acc, /*reuse_a=*/false, /*reuse_b=*/false);
    }

    // C/D layout: VGPR r -> M = r + 8*half, N = lane&15
    const int n = tileN * 16 + (lane & 15);
    #pragma unroll
    for (int r = 0; r < 8; ++r) {
        C[(size_t)(tileM * 16 + r + half * 8) * N + n] = acc[r];
    }
}

// ---------------------------------------------------------------------------
// Zero a float buffer (float4-vectorized; count must be a multiple of 4)
// ---------------------------------------------------------------------------
__global__ __launch_bounds__(256) void gcn_zero_f4(float4* __restrict__ p, long long n4) {
    long long i = (long long)blockIdx.x * blockDim.x + threadIdx.x;
    if (i < n4) p[i] = make_float4(0.f, 0.f, 0.f, 0.f);
}

// ---------------------------------------------------------------------------
// Edge scatter: for each edge e and channel group g (4 channels):
//   out[dst[e]][4g..4g+3] += H[src[e]][4g..4g+3] * dinv[src]*dinv[dst]
// groups = C/4 (power of two: 32 or 16), gshift = log2(groups)
// ---------------------------------------------------------------------------
__global__ __launch_bounds__(256) void gcn_edge_scatter(
    const long long* __restrict__ src, const long long* __restrict__ dst,
    const float* __restrict__ dinv, const float* __restrict__ H,
    float* __restrict__ out, long long total, int gshift, int gmask, int C) {
    long long t = (long long)blockIdx.x * blockDim.x + threadIdx.x;
    if (t >= total) return;
    long long e = t >> gshift;
    int g = (int)(t & gmask);

    int s = (int)src[e];
    int d = (int)dst[e];
    float w = dinv[s] * dinv[d];

    float4 v = *((const float4*)(H + (size_t)s * C) + g);
    float* op = out + (size_t)d * C + g * 4;
    atomicAdd(op + 0, v.x * w);
    atomicAdd(op + 1, v.y * w);
    atomicAdd(op + 2, v.z * w);
    atomicAdd(op + 3, v.w * w);
}

// ---------------------------------------------------------------------------
// Finalize: acc[n][c] = (acc[n][c] + H[n][c]*dinv[n]^2 + bias[c]) , opt. ReLU
// (self-loop message + bias, matching reference concat([edges, loops]))
// ---------------------------------------------------------------------------
__global__ __launch_bounds__(256) void gcn_finalize(
    const float* __restrict__ H, const float* __restrict__ dinv,
    const float* __restrict__ bias, float* __restrict__ acc,
    long long total, int C, int doRelu) {
    long long i = (long long)blockIdx.x * blockDim.x + threadIdx.x;
    if (i >= total) return;
    int node = (int)(i / C);
    int c = (int)(i % C);
    float di = dinv[node];
    float v = acc[i] + H[i] * di * di + bias[c];
    acc[i] = doRelu ? fmaxf(v, 0.0f) : v;
}

// ---------------------------------------------------------------------------
// Launch
// ---------------------------------------------------------------------------
extern "C" void kernel_launch(void* const* d_in, const int* in_sizes, int n_in,
                              void* d_out, int out_size, void* d_ws, size_t ws_size,
                              hipStream_t stream) {
    const float*     x   = (const float*)d_in[0];
    const long long* ei  = (const long long*)d_in[1];   // int64, shape (2, E) flat
    const float*     W1  = (const float*)d_in[2];
    const float*     b1  = (const float*)d_in[3];
    const float*     W2  = (const float*)d_in[4];
    const float*     b2  = (const float*)d_in[5];
    float*           out = (float*)d_out;               // [N_NODES, OUT_CH]

    const long long* src = ei;              // edge_index[0]
    const long long* dst = ei + N_EDGES;    // edge_index[1]

    // Workspace carving (floats)
    float* dinv = (float*)d_ws;                                 // N_NODES
    float* H1   = dinv + ((N_NODES + 255) & ~255);              // N*128 : X@W1
    float* A1   = H1 + (size_t)N_NODES * HID_CH;                // N*128 : aggregated layer1
    float* H2   = A1 + (size_t)N_NODES * HID_CH;                // N*64  : A1@W2

    const int T = 256;

    // 1) degrees -> dinv
    gcn_deg_init<<<(N_NODES + T - 1) / T, T, 0, stream>>>(dinv, N_NODES);
    gcn_deg_edges<<<(N_EDGES + T - 1) / T, T, 0, stream>>>(dst, dinv, N_EDGES);
    gcn_deg_to_dinv<<<(N_NODES + T - 1) / T, T, 0, stream>>>(dinv, N_NODES);

    // 2) H1 = X @ W1   (M=100000, K=128, N=128) — WMMA fp32
    {
        dim3 grid(N_NODES / 16, HID_CH / 64);  // 4 column-tiles per block (4 waves)
        gcn_gemm_wmma_f32<<<grid, 128, 0, stream>>>(x, W1, H1, N_NODES, IN_CH, HID_CH);
    }

    // 3) A1 = 0 ; scatter edges ; finalize (+self loop, +b1, ReLU) in place
    {
        long long n4 = (long long)N_NODES * HID_CH / 4;
        gcn_zero_f4<<<(unsigned)((n4 + T - 1) / T), T, 0, stream>>>((float4*)A1, n4);

        long long total = (long long)N_EDGES * (HID_CH / 4);
        gcn_edge_scatter<<<(unsigned)((total + T - 1) / T), T, 0, stream>>>(
            src, dst, dinv, H1, A1, total, /*gshift=*/5, /*gmask=*/31, HID_CH);

        long long tf = (long long)N_NODES * HID_CH;
        gcn_finalize<<<(unsigned)((tf + T - 1) / T), T, 0, stream>>>(
            H1, dinv, b1, A1, tf, HID_CH, /*doRelu=*/1);
    }

    // 4) H2 = A1 @ W2   (M=100000, K=128, N=64) — WMMA fp32
    {
        dim3 grid(N_NODES / 16, OUT_CH / 64);  // 64/64 = 1 block-col, 4 waves
        gcn_gemm_wmma_f32<<<grid, 128, 0, stream>>>(A1, W2, H2, N_NODES, HID_CH, OUT_CH);
    }

    // 5) out = 0 ; scatter edges ; finalize (+self loop, +b2, no ReLU)
    {
        long long n4 = (long long)N_NODES * OUT_CH / 4;
        gcn_zero_f4<<<(unsigned)((n4 + T - 1) / T), T, 0, stream>>>((float4*)out, n4);

        long long total = (long long)N_EDGES * (OUT_CH / 4);
        gcn_edge_scatter<<<(unsigned)((total + T - 1) / T), T, 0, stream>>>(
            src, dst, dinv, H2, out, total, /*gshift=*/4, /*gmask=*/15, OUT_CH);

        long long tf = (long long)N_NODES * OUT_CH;
        gcn_finalize<<<(unsigned)((tf + T - 1) / T), T, 0, stream>>>(
            H2, dinv, b2, out, tf, OUT_CH, /*doRelu=*/0);
    }
}